// MultiHeadAttention_71339406786720
// MI455X (gfx1250) — compile-verified
//
#include <hip/hip_runtime.h>

typedef float v2f __attribute__((ext_vector_type(2)));
typedef float v4f __attribute__((ext_vector_type(4)));
typedef float v8f __attribute__((ext_vector_type(8)));

#define BB   2
#define NN   10000
#define KNB  32        // neighbors per position
#define DM   256       // d_model
#define DK   64        // d_k
#define DK1  65        // padded LDS row (bank-conflict-free)
#define BN   (BB * NN) // 20000 positions

#define WAVES_PER_BLOCK 4
#define BLOCK (WAVES_PER_BLOCK * 32)
#define GRID  512

#define NKT   (DM / 4)          // 64 K-steps of 4
#define NFRAG (NKT * 4)         // 256 B-fragments (4 e-tiles each step)

// ---- pack w_ks [64 x 256] into WMMA-B fragment order: one v2f per lane ----
// Fragment f = ktile*4 + nt; lane L holds B[kk = 2*(L>>4) + {0,1}][nl = L&15]
// i.e. w_ks[e = nt*16 + (L&15)][d = 4*ktile + 2*(L>>4) + {0,1}], stored
// contiguously so the main loop issues a single coalesced global_load_b64.
__global__ void wks_pack_kernel(const float* __restrict__ w_ks,
                                v2f* __restrict__ wpk) {
    int tid = blockIdx.x * blockDim.x + threadIdx.x;  // 0 .. 8191
    int lane  = tid & 31;
    int nt    = (tid >> 5) & 3;
    int ktile = tid >> 7;
    int e  = nt * 16 + (lane & 15);
    int dA = ktile * 4 + ((lane >> 4) << 1);
    v2f p;
    p.x = w_ks[(size_t)e * DM + dA];
    p.y = w_ks[(size_t)e * DM + dA + 1];
    wpk[tid] = p;
}

// ---- main fused attention kernel: one wave per position ----
__global__ __launch_bounds__(BLOCK) void mha_fused_kernel(
    const float* __restrict__ q,     // [BN, 256]
    const float* __restrict__ kt,    // [BN, 32, 256]
    const float* __restrict__ v,     // [BN, 32, 64]
    const float* __restrict__ w_qs,  // [64, 256]
    const v2f*   __restrict__ wpk,   // packed B fragments (or null)
    const float* __restrict__ w_ks,  // raw fallback
    int packed,
    float* __restrict__ out,         // [BN, 64]
    float* __restrict__ q2)          // [BN, 256]
{
    __shared__ float kp_lds[WAVES_PER_BLOCK][KNB * DK1];  // 32x64 padded per wave
    __shared__ float q1_lds[WAVES_PER_BLOCK][DK];
    __shared__ float attn_lds[WAVES_PER_BLOCK][KNB];

    const int lane   = threadIdx.x & 31;
    const int wave   = threadIdx.x >> 5;
    const int gwave  = blockIdx.x * WAVES_PER_BLOCK + wave;
    const int nwaves = gridDim.x * WAVES_PER_BLOCK;

    // WMMA f32 16x16x4 A-fragment addressing: lane L holds M = L&15,
    // K-pair (2*(L>>4), 2*(L>>4)+1) in VGPRs {0,1}  -> one float2 load.
    const int m_lo  = lane & 15;
    const int khalf = (lane >> 4) << 1;   // 0 or 2

    for (int idx = gwave; idx < BN; idx += nwaves) {
        const float* Kn = kt + (size_t)idx * (KNB * DM);
        const float* Qn = q  + (size_t)idx * DM;
        const float* Vn = v  + (size_t)idx * (KNB * DK);

        // Prefetch first line of each row of the *next* K tile this wave will use.
        if (idx + nwaves < BN) {
            __builtin_prefetch(kt + (size_t)(idx + nwaves) * (KNB * DM) + lane * DM, 0, 1);
        }

        // ============ kp = K_n(32x256) x W_ks^T(256x64) via WMMA ============
        v8f acc[2][4] = {};   // [m-tile][e-tile] -> 32x64 fp32 output
        if (packed) {
            #pragma unroll 4
            for (int ktile = 0; ktile < NKT; ++ktile) {
                const int dA = ktile * 4 + khalf;
                v2f a0 = *(const v2f*)(Kn + (size_t)m_lo * DM + dA);
                v2f a1 = *(const v2f*)(Kn + (size_t)(m_lo + 16) * DM + dA);
                const v2f* wrow = wpk + ((size_t)(ktile * 4) << 5) + lane;
                #pragma unroll
                for (int nt = 0; nt < 4; ++nt) {
                    v2f bf = wrow[nt << 5];   // one coalesced b64 per fragment
                    acc[0][nt] = __builtin_amdgcn_wmma_f32_16x16x4_f32(
                        false, a0, false, bf, (short)0, acc[0][nt], false, false);
                    acc[1][nt] = __builtin_amdgcn_wmma_f32_16x16x4_f32(
                        false, a1, false, bf, (short)0, acc[1][nt], false, false);
                }
            }
        } else {
            #pragma unroll 4
            for (int ktile = 0; ktile < NKT; ++ktile) {
                const int dA = ktile * 4 + khalf;
                v2f a0 = *(const v2f*)(Kn + (size_t)m_lo * DM + dA);
                v2f a1 = *(const v2f*)(Kn + (size_t)(m_lo + 16) * DM + dA);
                #pragma unroll
                for (int nt = 0; nt < 4; ++nt) {
                    const int e = nt * 16 + m_lo;
                    v2f bf;
                    bf.x = w_ks[(size_t)e * DM + dA];
                    bf.y = w_ks[(size_t)e * DM + dA + 1];
                    acc[0][nt] = __builtin_amdgcn_wmma_f32_16x16x4_f32(
                        false, a0, false, bf, (short)0, acc[0][nt], false, false);
                    acc[1][nt] = __builtin_amdgcn_wmma_f32_16x16x4_f32(
                        false, a1, false, bf, (short)0, acc[1][nt], false, false);
                }
            }
        }

        // Spill kp to LDS (C layout: VGPR r, lane -> M = 16*mt + r + 8*(lane>>4))
        #pragma unroll
        for (int mt = 0; mt < 2; ++mt)
            #pragma unroll
            for (int nt = 0; nt < 4; ++nt)
                #pragma unroll
                for (int r = 0; r < 8; ++r) {
                    int m = mt * 16 + r + ((lane >> 4) << 3);
                    int e = nt * 16 + m_lo;
                    kp_lds[wave][m * DK1 + e] = acc[mt][nt][r];
                }

        // ============ q1 = W_qs x q_n  (each lane owns e=lane, e=lane+32) ====
        float q1a = 0.f, q1b = 0.f;
        #pragma unroll 4
        for (int d4 = 0; d4 < DM; d4 += 4) {
            v4f qv  = *(const v4f*)(Qn + d4);
            v4f wa  = *(const v4f*)(w_qs + (size_t)lane * DM + d4);
            v4f wbv = *(const v4f*)(w_qs + (size_t)(lane + 32) * DM + d4);
            q1a += qv.x * wa.x  + qv.y * wa.y  + qv.z * wa.z  + qv.w * wa.w;
            q1b += qv.x * wbv.x + qv.y * wbv.y + qv.z * wbv.z + qv.w * wbv.w;
        }
        q1_lds[wave][lane]      = q1a;
        q1_lds[wave][lane + 32] = q1b;
        asm volatile("s_wait_dscnt 0" ::: "memory");  // intra-wave LDS visibility

        // ============ scores + softmax over K=32 (lane k = lane) ============
        float s = 0.f;
        #pragma unroll 8
        for (int e = 0; e < DK; ++e)
            s += kp_lds[wave][lane * DK1 + e] * q1_lds[wave][e];
        // TEMPERATURE == 1.0

        float mx = s;
        #pragma unroll
        for (int off = 16; off > 0; off >>= 1)
            mx = fmaxf(mx, __shfl_xor(mx, off, 32));
        float p = __expf(s - mx);
        float sum = p;
        #pragma unroll
        for (int off = 16; off > 0; off >>= 1)
            sum += __shfl_xor(sum, off, 32);
        attn_lds[wave][lane] = p / sum;
        asm volatile("s_wait_dscnt 0" ::: "memory");

        // ============ out = attn . V_n  (coalesced V stream) =================
        float o0 = 0.f, o1 = 0.f;
        #pragma unroll 8
        for (int kk = 0; kk < KNB; ++kk) {
            float ak = attn_lds[wave][kk];
            o0 += ak * Vn[kk * DK + lane];
            o1 += ak * Vn[kk * DK + lane + 32];
        }
        out[(size_t)idx * DK + lane]      = o0;
        out[(size_t)idx * DK + lane + 32] = o1;

        // ============ q2 = q (flat copy, float4) ============================
        const v4f* Q4 = (const v4f*)Qn;
        v4f* O4 = (v4f*)(q2 + (size_t)idx * DM);
        O4[lane]      = Q4[lane];
        O4[lane + 32] = Q4[lane + 32];
    }
}

extern "C" void kernel_launch(void* const* d_in, const int* in_sizes, int n_in,
                              void* d_out, int out_size, void* d_ws, size_t ws_size,
                              hipStream_t stream) {
    const float* q    = (const float*)d_in[0];
    const float* k    = (const float*)d_in[1];
    const float* v    = (const float*)d_in[2];
    const float* w_qs = (const float*)d_in[3];
    const float* w_ks = (const float*)d_in[4];

    float* out = (float*)d_out;                 // [BN,64]
    float* q2  = out + (size_t)BN * DK;         // [BN,256]

    // Pre-pack w_ks into WMMA-B fragment order (one v2f per lane per fragment).
    const v2f* wpk = nullptr;
    int packed = 0;
    if (ws_size >= (size_t)NFRAG * 32 * sizeof(v2f)) {   // 64 KB
        v2f* wpk_ws = (v2f*)d_ws;
        hipLaunchKernelGGL(wks_pack_kernel, dim3(32), dim3(256), 0, stream,
                           w_ks, wpk_ws);
        wpk = wpk_ws;
        packed = 1;
    }

    hipLaunchKernelGGL(mha_fused_kernel, dim3(GRID), dim3(BLOCK), 0, stream,
                       q, k, v, w_qs, wpk, w_ks, packed, out, q2);
}